// CapacityBasedRouter_43508018708618
// MI455X (gfx1250) — compile-verified
//
#include <hip/hip_runtime.h>
#include <hip/hip_bf16.h>
#include <math.h>

// ---------------------------------------------------------------------------
// Capacity-based MoE router for MI455X (gfx1250, wave32, WMMA).
//   N=16384 tokens, D=4096, E=8 experts, top-k=2, capacity=2048.
// Phase 1 (GEMM, HBM-bound ~11us floor @ 23.3 TB/s): v_wmma_f32_16x16x4_f32,
//   one wave per 16-token tile; b128 loads feed 2 WMMAs each via K-permuted
//   fragments; fused softmax/top2/z-loss/histograms.
// Phase 2/3/4: per-expert histogram threshold -> capacity mask -> losses.
// ---------------------------------------------------------------------------

typedef __attribute__((ext_vector_type(2))) float v2f;
typedef __attribute__((ext_vector_type(4))) float v4f;
typedef __attribute__((ext_vector_type(8))) float v8f;

#define N_TOKENS   16384
#define D_MODEL    4096
#define NUM_EXP    8
#define TOP_K      2
#define CAPACITY   2048
#define N_ASSIGN   (N_TOKENS * TOP_K)      // 32768
#define N_BINS     1024

// d_out layout (floats):
//   [0 .. 32767]       top_k_indices (as float)
//   [32768 .. 65535]   top_k_probs
//   [65536]            load_balance_loss
//   [65537]            router_z_loss
//   [65538 .. 65545]   expert_usage
#define OUT_PROB_OFF  N_ASSIGN
#define OUT_LBL_OFF   (2 * N_ASSIGN)
#define OUT_ZL_OFF    (2 * N_ASSIGN + 1)
#define OUT_USE_OFF   (2 * N_ASSIGN + 2)

// workspace byte offsets
#define WS_TOPIDX_OFF  0                         // int[32768]
#define WS_TOPPRB_OFF  (N_ASSIGN * 4)            // float[32768]
#define WS_HIST_OFF    (2 * N_ASSIGN * 4)        // int[8*1024]
#define WS_MISC_OFF    (WS_HIST_OFF + NUM_EXP * N_BINS * 4) // tb[8],slots[8],slotctr[8],usage[8]
#define WS_ZSUM_OFF    (WS_MISC_OFF + 32 * 4)    // float[1]
#define WS_ZERO_LEN    (NUM_EXP * N_BINS * 4 + 32 * 4 + 4)

// ---------------------------------------------------------------------------
// Kernel 1: gate GEMM via WMMA + fused per-token routing.
// Block = 256 threads (8 waves), each wave owns one 16-token tile.
//
// Fragment trick: the K reduction is order-agnostic, so a WMMA's 4 K-columns
// need not be consecutive. Each lane loads float4 at column kk + khalf*4
// (lanes 0-15: kk..kk+3, lanes 16-31: kk+4..kk+7). Then:
//   WMMA #1 uses (v0,v1) -> K-set {kk, kk+1, kk+4, kk+5}
//   WMMA #2 uses (v2,v3) -> K-set {kk+2, kk+3, kk+6, kk+7}
// A and B share the mapping, so the dot products are exact.
// ---------------------------------------------------------------------------
__global__ __launch_bounds__(256) void router_gemm_topk_kernel(
    const float* __restrict__ x,        // [N, D]
    const float* __restrict__ gate_w,   // [E, D]
    int*   __restrict__ top_idx,        // [N*2]
    float* __restrict__ top_prob,       // [N*2]
    int*   __restrict__ hist,           // [E, 1024]
    float* __restrict__ zsum)           // [1]
{
    __shared__ float sbuf[8][16][NUM_EXP];   // per-wave logit transpose buffer

    const int lane = threadIdx.x & 31;
    const int wave = threadIdx.x >> 5;
    const int tile = blockIdx.x * 8 + wave;  // 16-token tile id

    const int row   = lane & 15;             // A: M row / B: N column
    const int khalf = lane >> 4;
    const int koff  = khalf * 4;             // float4 column base per lane

    const float* aptr = x + (size_t)(tile * 16 + row) * D_MODEL + koff;
    const bool   bval = (row < NUM_EXP);     // N columns 8..15 are zero padding
    // Clamped row: always a valid address; padding zeroed branchlessly below.
    const float* bptr = gate_w + (size_t)(row & 7) * D_MODEL + koff;

    const v4f vzero = {0.0f, 0.0f, 0.0f, 0.0f};
    v8f c = {};                              // 16x16 f32 accumulator (8 VGPRs)

#pragma unroll 4
    for (int kk = 0; kk < D_MODEL; kk += 8) {
        v4f a = *(const v4f*)(aptr + kk);            // global_load_b128
        v4f b = *(const v4f*)(bptr + kk);            // global_load_b128 (L2-hot)
        b = bval ? b : vzero;                        // v_cndmask, no branches
        c = __builtin_amdgcn_wmma_f32_16x16x4_f32(
                false, a.lo, false, b.lo, (short)0, c, false, false);
        c = __builtin_amdgcn_wmma_f32_16x16x4_f32(
                false, a.hi, false, b.hi, (short)0, c, false, false);
    }

    // C fragment: VGPR r -> M = r + 8*khalf, N = lane&15.  Keep N<8.
    if (row < NUM_EXP) {
#pragma unroll
        for (int r = 0; r < 8; ++r) {
            sbuf[wave][r + (khalf << 3)][row] = c[r];
        }
    }
    __syncthreads();

    // Lanes 0..15: one token each -> softmax / top-2 / z-loss / histogram.
    if (lane < 16) {
        const int t = tile * 16 + lane;
        float v[NUM_EXP];
#pragma unroll
        for (int e = 0; e < NUM_EXP; ++e) v[e] = sbuf[wave][lane][e];

        float mx = v[0];
#pragma unroll
        for (int e = 1; e < NUM_EXP; ++e) mx = fmaxf(mx, v[e]);
        float s = 0.0f;
#pragma unroll
        for (int e = 0; e < NUM_EXP; ++e) s += __expf(v[e] - mx);
        float lse = mx + __logf(s);
        atomicAdd(zsum, lse * lse);

        // top-2 (strict > keeps first index on ties, matching lax.top_k)
        float b1 = -3.0e38f, b2 = -3.0e38f;
        int   i1 = 0, i2 = 0;
#pragma unroll
        for (int e = 0; e < NUM_EXP; ++e) {
            float val = v[e];
            if (val > b1)      { b2 = b1; i2 = i1; b1 = val; i1 = e; }
            else if (val > b2) { b2 = val; i2 = e; }
        }
        float p1 = 1.0f / (1.0f + __expf(b2 - b1));   // softmax over top-2
        float p2 = 1.0f - p1;                         // renormalize: no-op

        top_idx [t * 2 + 0] = i1;
        top_idx [t * 2 + 1] = i2;
        top_prob[t * 2 + 0] = p1;
        top_prob[t * 2 + 1] = p2;

        int bin1 = min(N_BINS - 1, (int)(p1 * (float)N_BINS));
        int bin2 = min(N_BINS - 1, (int)(p2 * (float)N_BINS));
        atomicAdd(&hist[i1 * N_BINS + bin1], 1);
        atomicAdd(&hist[i2 * N_BINS + bin2], 1);
    }
}

// ---------------------------------------------------------------------------
// Kernel 2: per-expert threshold bin (suffix scan of 1024-bin histogram).
// ---------------------------------------------------------------------------
__global__ void expert_threshold_kernel(const int* __restrict__ hist,
                                        int* __restrict__ tb,
                                        int* __restrict__ slots)
{
    int e = threadIdx.x;
    if (e >= NUM_EXP) return;
    int cum = 0, t = -1, s = 0;
    for (int b = N_BINS - 1; b >= 0; --b) {
        int h = hist[e * N_BINS + b];
        if (cum + h >= CAPACITY) { t = b; s = CAPACITY - cum; break; }
        cum += h;
    }
    tb[e] = t;        // -1 => expert under capacity: keep all
    slots[e] = s;     // slots available in the threshold bin
}

// ---------------------------------------------------------------------------
// Kernel 3: apply capacity mask per assignment; emit indices/probs; usage.
// ---------------------------------------------------------------------------
__global__ __launch_bounds__(256) void apply_capacity_kernel(
    const int*   __restrict__ top_idx,
    const float* __restrict__ top_prob,
    const int*   __restrict__ tb,
    const int*   __restrict__ slots,
    int*         __restrict__ slotctr,
    int*         __restrict__ usage,
    float*       __restrict__ out)
{
    int i = blockIdx.x * blockDim.x + threadIdx.x;
    if (i >= N_ASSIGN) return;

    int   e = top_idx[i];
    float p = top_prob[i];
    int bin = min(N_BINS - 1, (int)(p * (float)N_BINS));

    int  t = tb[e];
    bool keep;
    if (t < 0)           keep = true;
    else if (bin > t)    keep = true;
    else if (bin == t)   keep = (atomicAdd(&slotctr[e], 1) < slots[e]);
    else                 keep = false;

    int fi = keep ? e : 0;                 // dropped -> index 0 (reference quirk)
    out[i]                = (float)fi;
    out[OUT_PROB_OFF + i] = keep ? p : 0.0f;
    atomicAdd(&usage[fi], 1);              // usage counts masked indices (incl. 0)
}

// ---------------------------------------------------------------------------
// Kernel 4: losses + expert_usage output.
// ---------------------------------------------------------------------------
__global__ void finalize_kernel(const int* __restrict__ usage,
                                const float* __restrict__ zsum,
                                float* __restrict__ out)
{
    int t = threadIdx.x;
    if (t < NUM_EXP) out[OUT_USE_OFF + t] = (float)usage[t];
    if (t == 0) {
        const float ideal = (float)(N_TOKENS * TOP_K / NUM_EXP);  // 4096
        float acc = 0.0f;
        for (int e = 0; e < NUM_EXP; ++e) {
            float d = (float)usage[e] - ideal;
            acc += d * d;
        }
        out[OUT_LBL_OFF] = acc / (float)NUM_EXP;
        out[OUT_ZL_OFF]  = zsum[0] / (float)N_TOKENS;
    }
}

// ---------------------------------------------------------------------------
extern "C" void kernel_launch(void* const* d_in, const int* in_sizes, int n_in,
                              void* d_out, int out_size, void* d_ws, size_t ws_size,
                              hipStream_t stream)
{
    const float* x      = (const float*)d_in[0];   // [16384, 4096] f32
    const float* gate_w = (const float*)d_in[1];   // [8, 4096] f32
    float* out = (float*)d_out;

    char* ws = (char*)d_ws;
    int*   top_idx  = (int*)  (ws + WS_TOPIDX_OFF);
    float* top_prob = (float*)(ws + WS_TOPPRB_OFF);
    int*   hist     = (int*)  (ws + WS_HIST_OFF);
    int*   tb       = (int*)  (ws + WS_MISC_OFF);
    int*   slots    = tb + NUM_EXP;
    int*   slotctr  = tb + 2 * NUM_EXP;
    int*   usage    = tb + 3 * NUM_EXP;
    float* zsum     = (float*)(ws + WS_ZSUM_OFF);

    // Zero hist / counters / zsum (graph-capture-legal memset node).
    hipMemsetAsync(ws + WS_HIST_OFF, 0, WS_ZERO_LEN, stream);

    // Phase 1: 1024 tiles of 16 tokens; 8 waves (tiles) per 256-thread block.
    router_gemm_topk_kernel<<<N_TOKENS / (16 * 8), 256, 0, stream>>>(
        x, gate_w, top_idx, top_prob, hist, zsum);

    // Phase 2: per-expert capacity threshold.
    expert_threshold_kernel<<<1, NUM_EXP, 0, stream>>>(hist, tb, slots);

    // Phase 3: keep/drop each of the 32768 assignments.
    apply_capacity_kernel<<<N_ASSIGN / 256, 256, 0, stream>>>(
        top_idx, top_prob, tb, slots, slotctr, usage, out);

    // Phase 4: losses + usage output.
    finalize_kernel<<<1, 32, 0, stream>>>(usage, zsum, out);
}